// DimWiseFusion_58506044506609
// MI455X (gfx1250) — compile-verified
//
#include <hip/hip_runtime.h>
#include <stdint.h>

// ---------------- problem constants ----------------
constexpr int  kE0   = 3200000;
constexpr int  kEN   = 3200000;
constexpr int  kN    = 6400000;              // total edges
constexpr int  kTile = 4096;                 // keys per sort tile (256 thr x 16)
constexpr int  kNB   = (kN + kTile - 1) / kTile;   // 1563 tiles
constexpr int  kNPad = kNB * kTile;          // 6402048 (padded key count)
constexpr int  kFlat = 256 * kNB;            // 400128 histogram entries / pass
constexpr int  kNCh  = (kFlat + 4095) / 4096;      // 98 scan chunks
constexpr unsigned long long kSent = 1ULL << 34;   // dropped-edge sentinel
constexpr unsigned long long kPadK = 0xFFFFFFFFFFULL; // tile padding (40 bits)

typedef float v2f __attribute__((ext_vector_type(2)));
typedef float v8f __attribute__((ext_vector_type(8)));

// ---------------- threefry2x32 (JAX-compatible) ----------------
__device__ __forceinline__ uint32_t rotl32(uint32_t x, int r) {
  return (x << r) | (x >> (32 - r));
}

__device__ void threefry2x32(uint32_t k0, uint32_t k1, uint32_t x0, uint32_t x1,
                             uint32_t& y0, uint32_t& y1) {
  const uint32_t ks2 = k0 ^ k1 ^ 0x1BD11BDAu;
  x0 += k0; x1 += k1;
#define TF_R(r) { x0 += x1; x1 = rotl32(x1, (r)); x1 ^= x0; }
  TF_R(13) TF_R(15) TF_R(26) TF_R(6)   x0 += k1;  x1 += ks2 + 1u;
  TF_R(17) TF_R(29) TF_R(16) TF_R(24)  x0 += ks2; x1 += k0  + 2u;
  TF_R(13) TF_R(15) TF_R(26) TF_R(6)   x0 += k0;  x1 += k1  + 3u;
  TF_R(17) TF_R(29) TF_R(16) TF_R(24)  x0 += k1;  x1 += ks2 + 4u;
  TF_R(13) TF_R(15) TF_R(26) TF_R(6)   x0 += ks2; x1 += k0  + 5u;
#undef TF_R
  y0 = x0; y1 = x1;
}

// ---------------- WMMA-based exclusive scan of 256 values ----------------
// Each of 256 threads contributes one f32 (an exact small integer). Wave 0
// computes the scan as D = L_tri x X with 4 accumulating v_wmma_f32_16x16x4_f32
// (column-major mapping: element i -> row i%16, col i/16). Exact for integer
// totals < 2^24 (our max is 6.4M).
__device__ float block_scan_256(float v, float* total_out) {
  __shared__ float xs[256];
  __shared__ float ss[256];
  __shared__ float tots[16];
  __shared__ float totalsh;
  const int tid = threadIdx.x;
  xs[tid] = v;
  __syncthreads();
  if (tid < 32) {                        // wave 0, EXEC all ones
    const int l    = tid;
    const int half = l >> 4;             // 0: lanes 0-15, 1: lanes 16-31
    const int mn   = l & 15;             // M for A-layout, N for B/C-layout
    v8f acc = {0.f, 0.f, 0.f, 0.f, 0.f, 0.f, 0.f, 0.f};
#pragma unroll
    for (int c = 0; c < 4; ++c) {
      const int k0 = 4 * c + 2 * half;   // K held in VGPR0 (ISA 16x4 A layout)
      const int k1 = k0 + 1;             // K held in VGPR1
      v2f a, b;
      a.x = (k0 <= mn) ? 1.f : 0.f;      // inclusive lower-triangular L
      a.y = (k1 <= mn) ? 1.f : 0.f;
      b.x = xs[mn * 16 + k0];            // X[k][n], column-major storage
      b.y = xs[mn * 16 + k1];
      acc = __builtin_amdgcn_wmma_f32_16x16x4_f32(
          false, a, false, b, (short)0, acc, false, false);
    }
    // column totals live in D[15][n]: lanes 16-31, VGPR 7
    if (half) tots[mn] = acc[7];
    asm volatile("s_wait_dscnt 0" ::: "memory");
    float cb = 0.f;
    for (int k = 0; k < mn; ++k) cb += tots[k];  // cross-column offset
#pragma unroll
    for (int vv = 0; vv < 8; ++vv) {
      const int m = vv + 8 * half;       // C-layout row
      const int i = mn * 16 + m;         // flat scan index
      ss[i] = acc[vv] - xs[i] + cb;      // exclusive prefix
    }
    if (l == 0) {
      float t = 0.f;
      for (int k = 0; k < 16; ++k) t += tots[k];
      totalsh = t;
    }
  }
  __syncthreads();
  *total_out = totalsh;
  return ss[tid];
}

// ---------------- kernel: fill output with -1 ----------------
__global__ void fill_out(float* __restrict__ out, size_t n) {
  size_t i = (size_t)blockIdx.x * blockDim.x + threadIdx.x;
  if (i < n) out[i] = -1.0f;
}

// ---------------- kernel: build 34-bit keys (+ threefry bernoulli) --------
__global__ void __launch_bounds__(256) build_codes(
    const long long* __restrict__ A0, const long long* __restrict__ An,
    const float* __restrict__ p, unsigned long long* __restrict__ keys) {
  const long long i = (long long)blockIdx.x * 256 + threadIdx.x;
  unsigned long long key;
  if (i >= kN) {
    key = kPadK;                              // tile padding, sorts last
  } else if (i >= kE0) {
    const long long j = i - kE0;
    const unsigned long long s = (unsigned long long)An[j];
    const unsigned long long d = (unsigned long long)An[(long long)kEN + j];
    key = (s << 17) | d;
  } else {
    // jax.random.bernoulli(key(42), sigmoid(p), (E0,))
    const uint32_t half = (uint32_t)(kE0 / 2);
    uint32_t x0, x1; bool hi;
    if ((uint32_t)i < half) { x0 = (uint32_t)i;        x1 = (uint32_t)i + half; hi = false; }
    else                    { x0 = (uint32_t)i - half; x1 = (uint32_t)i;        hi = true;  }
    uint32_t y0, y1;
    threefry2x32(0u, 42u, x0, x1, y0, y1);
    const uint32_t bits = hi ? y1 : y0;
    const float u = __uint_as_float((bits >> 9) | 0x3F800000u) - 1.0f;
    const float keep_prob = 1.0f / (1.0f + expf(-p[0]));
    const unsigned long long s = (unsigned long long)A0[i];
    const unsigned long long d = (unsigned long long)A0[(long long)kE0 + i];
    key = (u < keep_prob) ? ((s << 17) | d) : kSent;
  }
  keys[i] = key;
}

// ---------------- kernel: per-tile 256-bin histogram ----------------
__global__ void __launch_bounds__(256) hist_kernel(
    const unsigned long long* __restrict__ in, uint32_t* __restrict__ hist,
    int shift) {
  __shared__ uint32_t cnt[256];
  const int tid = threadIdx.x, b = blockIdx.x;
  cnt[tid] = 0;
  __syncthreads();
#pragma unroll
  for (int j = 0; j < 16; ++j) {
    const unsigned long long key = in[(size_t)b * kTile + j * 256 + tid];
    atomicAdd(&cnt[(unsigned)(key >> shift) & 255u], 1u);
  }
  __syncthreads();
  hist[(size_t)tid * kNB + b] = cnt[tid];     // [digit][block] layout
}

// ---------------- kernel: exclusive scan of 4096-entry chunks (WMMA) ------
__global__ void __launch_bounds__(256) scan_chunks(
    uint32_t* __restrict__ data, uint32_t* __restrict__ chunksum, int n) {
  const int tid = threadIdx.x, c = blockIdx.x;
  const int base = c * 4096 + tid * 16;
  float vals[16];
#pragma unroll
  for (int k = 0; k < 16; ++k) {
    const int f = base + k;
    vals[k] = (f < n) ? (float)data[f] : 0.f;
  }
  float s = 0.f;
#pragma unroll
  for (int k = 0; k < 16; ++k) s += vals[k];
  float tot;
  float r = block_scan_256(s, &tot);
#pragma unroll
  for (int k = 0; k < 16; ++k) {
    const int f = base + k;
    if (f < n) data[f] = (uint32_t)r;         // in-place exclusive scan
    r += vals[k];
  }
  if (tid == 0) chunksum[c] = (uint32_t)tot;
}

// ---------------- kernel: single-block looped scan (WMMA per tile) --------
__global__ void __launch_bounds__(256) scan_small(uint32_t* __restrict__ a, int n) {
  const int tid = threadIdx.x;
  float carry = 0.f;
  const int niter = (n + 255) / 256;
  for (int t = 0; t < niter; ++t) {
    const int idx = t * 256 + tid;
    const float v = (idx < n) ? (float)a[idx] : 0.f;
    float tot;
    const float e = block_scan_256(v, &tot);
    if (idx < n) a[idx] = (uint32_t)(e + carry);
    carry += tot;
    __syncthreads();
  }
}

// ---------------- kernel: stable scatter (async-to-LDS staging) -----------
__global__ void __launch_bounds__(256) scatter_kernel(
    const unsigned long long* __restrict__ in,
    unsigned long long* __restrict__ out,
    const uint32_t* __restrict__ scanl, const uint32_t* __restrict__ choff,
    int shift) {
  __shared__ unsigned long long tk[kTile];    // 32 KB key tile
  __shared__ uint32_t wc[8 * 256];            // per-wave digit counts
  __shared__ uint32_t base[256];              // global digit bases for tile
  __shared__ uint32_t run[256];               // digits scattered so far
  const int tid = threadIdx.x, b = blockIdx.x;
  const int lane = tid & 31, w = tid >> 5;

  // Stage the tile into LDS via the CDNA5 async path (ASYNCcnt tracked).
#pragma unroll
  for (int j = 0; j < 16; ++j) {
    const int e = j * 256 + tid;
    // low 32 bits of the flat address of a __shared__ object are its LDS offset
    const unsigned lws = (unsigned)(unsigned long long)(uintptr_t)&tk[e];
    const unsigned long long gaddr =
        (unsigned long long)(uintptr_t)(in + (size_t)b * kTile + e);
    asm volatile("global_load_async_to_lds_b64 %0, %1, off"
                 :: "v"(lws), "v"(gaddr) : "memory");
  }
  {
    const int f = tid * kNB + b;              // flat hist position (digit=tid)
    base[tid] = scanl[f] + choff[f >> 12];
    run[tid] = 0;
  }
  asm volatile("s_wait_asynccnt 0" ::: "memory");
  __syncthreads();

  for (int j = 0; j < 16; ++j) {
#pragma unroll
    for (int z = 0; z < 8; ++z) wc[z * 256 + tid] = 0;
    __syncthreads();
    const unsigned long long key = tk[j * 256 + tid];
    const int d = (int)((key >> shift) & 255u);
    // wave32 peer matching over the 8 digit bits -> stable in-wave rank
    unsigned peers = 0xFFFFFFFFu;
#pragma unroll
    for (int bit = 0; bit < 8; ++bit) {
      const unsigned bal = __builtin_amdgcn_ballot_w32(((d >> bit) & 1) != 0);
      peers &= ((d >> bit) & 1) ? bal : ~bal;
    }
    const unsigned below = (1u << lane) - 1u;
    const int rank = __builtin_popcount(peers & below);
    if (rank == 0) wc[w * 256 + d] = (uint32_t)__builtin_popcount(peers);
    __syncthreads();
    uint32_t cross = 0;
    for (int ww = 0; ww < w; ++ww) cross += wc[ww * 256 + d];
    const uint32_t gpos = base[d] + run[d] + cross + (uint32_t)rank;
    out[gpos] = key;
    __syncthreads();
    uint32_t tsum = 0;
#pragma unroll
    for (int z = 0; z < 8; ++z) tsum += wc[z * 256 + tid];
    run[tid] += tsum;
    __syncthreads();
  }
}

// ---------------- kernels: dedup + compaction ----------------
__global__ void __launch_bounds__(256) flag_count(
    const unsigned long long* __restrict__ keys, uint32_t* __restrict__ csum) {
  const int tid = threadIdx.x, b = blockIdx.x;
  const long long base = (long long)b * kTile + tid * 16;
  unsigned long long prev = (base > 0) ? keys[base - 1] : ~0ULL;
  uint32_t c = 0;
#pragma unroll
  for (int k = 0; k < 16; ++k) {
    const unsigned long long cur = keys[base + k];
    c += (cur < kSent && cur != prev) ? 1u : 0u;
    prev = cur;
  }
  float tot;
  (void)block_scan_256((float)c, &tot);
  if (tid == 0) csum[b] = (uint32_t)tot;
}

__global__ void __launch_bounds__(256) compact_write(
    const unsigned long long* __restrict__ keys,
    const uint32_t* __restrict__ csum, float* __restrict__ out) {
  const int tid = threadIdx.x, b = blockIdx.x;
  const long long base = (long long)b * kTile + tid * 16;
  const unsigned long long prev0 = (base > 0) ? keys[base - 1] : ~0ULL;
  unsigned long long kv[16];
  unsigned long long prev = prev0;
  uint32_t c = 0;
#pragma unroll
  for (int k = 0; k < 16; ++k) {
    kv[k] = keys[base + k];
    c += (kv[k] < kSent && kv[k] != prev) ? 1u : 0u;
    prev = kv[k];
  }
  float tot;
  const float ex = block_scan_256((float)c, &tot);
  uint32_t g = csum[b] + (uint32_t)ex;
  prev = prev0;
#pragma unroll
  for (int k = 0; k < 16; ++k) {
    const unsigned long long cur = kv[k];
    if (cur < kSent && cur != prev) {
      out[g]              = (float)(unsigned)(cur >> 17);       // src
      out[(size_t)kN + g] = (float)(unsigned)(cur & 0x1FFFFu);  // dst
      ++g;
    }
    prev = cur;
  }
}

// ---------------- host launcher ----------------
extern "C" void kernel_launch(void* const* d_in, const int* in_sizes, int n_in,
                              void* d_out, int out_size, void* d_ws, size_t ws_size,
                              hipStream_t stream) {
  (void)in_sizes; (void)n_in; (void)out_size;
  const long long* A0 = (const long long*)d_in[0];
  const long long* An = (const long long*)d_in[1];
  const float*     p  = (const float*)d_in[3];
  float* out = (float*)d_out;

  char* ws = (char*)d_ws;
  size_t off = 0;
  auto alloc = [&](size_t bytes) -> void* {
    void* r = ws + off;
    off += (bytes + 255) & ~(size_t)255;
    return r;
  };
  unsigned long long* k0 = (unsigned long long*)alloc((size_t)kNPad * 8);
  unsigned long long* k1 = (unsigned long long*)alloc((size_t)kNPad * 8);
  uint32_t* hist     = (uint32_t*)alloc((size_t)kFlat * 4);
  uint32_t* chunksum = (uint32_t*)alloc((size_t)kNCh * 4);
  uint32_t* csum     = (uint32_t*)alloc((size_t)kNB * 4);
  if (off > ws_size) return;  // workspace too small; nothing safe to do

  fill_out<<<(2 * kN + 1023) / 1024, 1024, 0, stream>>>(out, (size_t)2 * kN);
  build_codes<<<kNPad / 256, 256, 0, stream>>>(A0, An, p, k0);

  unsigned long long* src = k0;
  unsigned long long* dst = k1;
  for (int pass = 0; pass < 5; ++pass) {      // 5 x 8 bits covers 40-bit keys
    const int shift = 8 * pass;
    hist_kernel <<<kNB, 256, 0, stream>>>(src, hist, shift);
    scan_chunks <<<kNCh, 256, 0, stream>>>(hist, chunksum, kFlat);
    scan_small  <<<1,    256, 0, stream>>>(chunksum, kNCh);
    scatter_kernel<<<kNB, 256, 0, stream>>>(src, dst, hist, chunksum, shift);
    unsigned long long* t = src; src = dst; dst = t;
  }
  // after 5 passes the sorted keys are in `src`
  flag_count   <<<kNB, 256, 0, stream>>>(src, csum);
  scan_small   <<<1,   256, 0, stream>>>(csum, kNB);
  compact_write<<<kNB, 256, 0, stream>>>(src, csum, out);
}